// MultiHeadAttention_65987877535893
// MI455X (gfx1250) — compile-verified
//
#include <hip/hip_runtime.h>
#include <cstdint>

typedef float v2f __attribute__((ext_vector_type(2)));
typedef float v8f __attribute__((ext_vector_type(8)));

#define S_LEN  2048
#define DMODEL 1024
#define NH     16
#define DH     64
#define BATCH  2
#define PSTR   20   // LDS row stride (floats) in attention: conflict-free & 8B-aligned

__device__ __forceinline__ v8f wmma_f32(v2f a, v2f b, v8f c) {
  // V_WMMA_F32_16X16X4_F32 : D = A(16x4) * B(4x16) + C(16x16), fp32
  return __builtin_amdgcn_wmma_f32_16x16x4_f32(
      /*neg_a=*/false, a, /*neg_b=*/false, b,
      /*c_mod=*/(short)0, c, /*reuse_a=*/false, /*reuse_b=*/false);
}

// Async DMA: 16 bytes global -> LDS, tracked by ASYNCcnt.
// Per ISA 10.2, the low 32 bits of a generic LDS pointer are the LDS byte offset.
__device__ __forceinline__ void async_copy_b128(const float* gsrc, float* ldst) {
  uint32_t loff = (uint32_t)(uintptr_t)ldst;
  asm volatile("global_load_async_to_lds_b128 %0, %1, off"
               :: "v"(loff), "v"(gsrc) : "memory");
}

// ---------------------------------------------------------------------------
// GEMM: C[M,1024] = A[M,1024] @ W[1024,1024]
// Block = 128 thr (4 waves), tile 64x128; wave tile 32x64 (2x4 WMMA tiles).
// W K-chunks (16x128) double-buffered in LDS via async-to-LDS DMA.
// permute=0: row-major store. permute=1: scatter to [B,H,S,Dh].
// ---------------------------------------------------------------------------
#define KCH 16                       // K-chunk
#define NCHUNK (DMODEL / KCH)        // 64
#define SEGS   (KCH * 128 / 4)       // 512 b128 segments per chunk
#define PT     (SEGS / 128)          // 4 async ops per thread per chunk

__global__ __launch_bounds__(128)
void gemm_kernel(const float* __restrict__ A, const float* __restrict__ W,
                 float* __restrict__ C, int permute) {
  __shared__ float wtile[2][KCH * 128];

  const int t    = threadIdx.x;
  const int lane = t & 31;
  const int wave = t >> 5;
  const int ln = lane & 15;
  const int lh = lane >> 4;                 // 0: K {0,1}/rows 0-7 ; 1: K {2,3}/rows 8-15
  const int m0 = blockIdx.y * 64 + (wave >> 1) * 32;
  const int nblk = blockIdx.x * 128;        // block N origin
  const int nw = (wave & 1) * 64;           // wave N origin within block

  v8f zero = {0.f,0.f,0.f,0.f,0.f,0.f,0.f,0.f};
  v8f acc[2][4];
#pragma unroll
  for (int mi = 0; mi < 2; ++mi)
#pragma unroll
    for (int ni = 0; ni < 4; ++ni) acc[mi][ni] = zero;

  const float* arow0 = A + (size_t)(m0 + ln) * DMODEL;
  const float* arow1 = A + (size_t)(m0 + 16 + ln) * DMODEL;
  const float* wsrc  = W + nblk;

  // issue async DMA of chunk c into buffer b: W[c*16 .. +16)[nblk .. +128)
  auto issue = [&](int c, int b) {
#pragma unroll
    for (int i = 0; i < PT; ++i) {
      const int seg = t + i * 128;          // 0..511
      const int row = seg >> 5;             // 128 floats/row = 32 b128 segs
      const int cs  = (seg & 31) * 4;
      async_copy_b128(wsrc + (size_t)(c * KCH + row) * DMODEL + cs,
                      &wtile[b][row * 128 + cs]);
    }
  };

  issue(0, 0);
  for (int c = 0; c < NCHUNK; ++c) {
    if (c + 1 < NCHUNK) {
      issue(c + 1, (c + 1) & 1);
      asm volatile("s_wait_asynccnt 4" ::: "memory");   // chunk c landed (our share)
    } else {
      asm volatile("s_wait_asynccnt 0" ::: "memory");
    }
    __syncthreads();                                    // everyone's share landed

    if (c + 2 < NCHUNK)                                 // prefetch A stream ahead
      __builtin_prefetch(arow0 + (c + 2) * KCH, 0, 1);

    const float* wb = &wtile[c & 1][0];
#pragma unroll
    for (int kk = 0; kk < KCH; kk += 4) {
      const int kd = kk + 2 * lh;
      v2f a0 = *(const v2f*)(arow0 + c * KCH + kd);
      v2f a1 = *(const v2f*)(arow1 + c * KCH + kd);
      const float* wr0 = wb + kd * 128 + nw + ln;
      const float* wr1 = wr0 + 128;
#pragma unroll
      for (int ni = 0; ni < 4; ++ni) {
        v2f b;
        b.x = wr0[ni * 16];
        b.y = wr1[ni * 16];
        acc[0][ni] = wmma_f32(a0, b, acc[0][ni]);
        acc[1][ni] = wmma_f32(a1, b, acc[1][ni]);
      }
    }
    __syncthreads();            // reads done before buffer (c&1) is overwritten
  }

#pragma unroll
  for (int mi = 0; mi < 2; ++mi)
#pragma unroll
    for (int ni = 0; ni < 4; ++ni)
#pragma unroll
      for (int r = 0; r < 8; ++r) {
        const int gm = m0 + mi * 16 + r + 8 * lh;       // row (b*S + s)
        const int gn = nblk + nw + ni * 16 + ln;        // col (h*Dh + d)
        const float val = acc[mi][ni][r];
        if (!permute) {
          C[(size_t)gm * DMODEL + gn] = val;
        } else {
          const int b = gm >> 11, s = gm & (S_LEN - 1);
          const int h = gn >> 6,  d = gn & (DH - 1);
          C[(((size_t)(b * NH + h)) * S_LEN + s) * DH + d] = val;
        }
      }
}

// ---------------------------------------------------------------------------
// In-place RoPE on Q and K, layout [B*H, S, Dh]; pairs (d, d+32), d<32.
// ---------------------------------------------------------------------------
__global__ __launch_bounds__(256)
void rope_kernel(float* __restrict__ Q, float* __restrict__ K,
                 const float* __restrict__ cosT, const float* __restrict__ sinT) {
  const int per = BATCH * NH * S_LEN * 32;   // 2,097,152 per tensor
  int tid = blockIdx.x * blockDim.x + threadIdx.x;
  float* X = (tid < per) ? Q : K;
  int r = (tid < per) ? tid : tid - per;
  const int d  = r & 31;
  const int s  = (r >> 5) & (S_LEN - 1);
  const int bh = r >> 16;                    // 32 * 2048 = 2^16
  const size_t base = ((size_t)bh * S_LEN + s) * DH;
  const float c  = cosT[s * 32 + d];
  const float sn = sinT[s * 32 + d];
  const float x1 = X[base + d];
  const float x2 = X[base + d + 32];
  X[base + d]      = x1 * c - x2 * sn;
  X[base + d + 32] = x2 * c + x1 * sn;
}

// ---------------------------------------------------------------------------
// Flash attention: one wave per 16-query tile, causal key loop in 16-wide tiles.
// ---------------------------------------------------------------------------
__global__ __launch_bounds__(256)
void attn_kernel(const float* __restrict__ Q, const float* __restrict__ K,
                 const float* __restrict__ V, float* __restrict__ AO) {
  __shared__ float pbuf[8][16 * PSTR];
  const int lane = threadIdx.x & 31;
  const int wave = threadIdx.x >> 5;
  const int ln = lane & 15;
  const int lh = lane >> 4;
  const int bh = blockIdx.y;
  const int b = bh >> 4, h = bh & 15;
  const int q0 = (blockIdx.x * 8 + wave) * 16;

  const float* Qb = Q + (size_t)bh * S_LEN * DH;
  const float* Kb = K + (size_t)bh * S_LEN * DH;
  const float* Vb = V + (size_t)bh * S_LEN * DH;

  // Preload Q tile as 16 A-fragments (pre-scaled by Dh^-0.5).
  const float scaling = 0.125f;
  v2f qa[16];
  {
    const float* qrow = Qb + (size_t)(q0 + ln) * DH + 2 * lh;
#pragma unroll
    for (int d4 = 0; d4 < 16; ++d4) {
      v2f t = *(const v2f*)(qrow + d4 * 4);
      qa[d4].x = t.x * scaling;
      qa[d4].y = t.y * scaling;
    }
  }

  v8f zero = {0.f,0.f,0.f,0.f,0.f,0.f,0.f,0.f};
  v8f o0 = zero, o1 = zero, o2 = zero, o3 = zero;   // 16 x 64 output accum
  float mrow[8], lrow[8];
#pragma unroll
  for (int r = 0; r < 8; ++r) { mrow[r] = -3.0e38f; lrow[r] = 0.f; }

  float* pw = &pbuf[wave][0];

  for (int j0 = 0; j0 <= q0; j0 += 16) {
    // --- scores: S = (Q*scale) @ K^T, 16x16 tile over Dh=64 ---
    v8f s = zero;
    const float* krow = Kb + (size_t)(j0 + ln) * DH + 2 * lh;
#pragma unroll
    for (int d4 = 0; d4 < 16; ++d4) {
      v2f kb = *(const v2f*)(krow + d4 * 4);
      s = wmma_f32(qa[d4], kb, s);
    }

    // --- online softmax (row reductions across 16-lane halves) ---
    const int j = j0 + ln;
    float p[8];
#pragma unroll
    for (int r = 0; r < 8; ++r) {
      const int qi = q0 + r + 8 * lh;
      float sv = s[r] + ((j > qi) ? -1.0e9f : 0.0f);   // causal mask
      float rm = sv;
      rm = fmaxf(rm, __shfl_xor(rm, 1));
      rm = fmaxf(rm, __shfl_xor(rm, 2));
      rm = fmaxf(rm, __shfl_xor(rm, 4));
      rm = fmaxf(rm, __shfl_xor(rm, 8));
      const float mold = mrow[r];
      const float mnew = fmaxf(mold, rm);
      const float corr = __expf(mold - mnew);
      const float pv   = __expf(sv - mnew);
      float rs = pv;
      rs += __shfl_xor(rs, 1);
      rs += __shfl_xor(rs, 2);
      rs += __shfl_xor(rs, 4);
      rs += __shfl_xor(rs, 8);
      mrow[r] = mnew;
      lrow[r] = lrow[r] * corr + rs;
      p[r] = pv;
      o0[r] *= corr; o1[r] *= corr; o2[r] *= corr; o3[r] *= corr;
    }

    // --- restage P (C-layout -> A-layout) through LDS ---
#pragma unroll
    for (int r = 0; r < 8; ++r)
      pw[(r + 8 * lh) * PSTR + ln] = p[r];
    asm volatile("s_wait_dscnt 0" ::: "memory");

    // --- O += P @ V (16 x 64), K-dim = 16 keys ---
#pragma unroll
    for (int kk = 0; kk < 16; kk += 4) {
      v2f pa = *(const v2f*)(pw + ln * PSTR + kk + 2 * lh);
      const int kr = j0 + kk + 2 * lh;
      const float* vrow = Vb + (size_t)kr * DH + ln;
      v2f vb;
      vb.x = vrow[0];       vb.y = vrow[DH];        o0 = wmma_f32(pa, vb, o0);
      vb.x = vrow[16];      vb.y = vrow[DH + 16];   o1 = wmma_f32(pa, vb, o1);
      vb.x = vrow[32];      vb.y = vrow[DH + 32];   o2 = wmma_f32(pa, vb, o2);
      vb.x = vrow[48];      vb.y = vrow[DH + 48];   o3 = wmma_f32(pa, vb, o3);
    }
    asm volatile("s_wait_dscnt 0" ::: "memory");  // pw reused next iteration
  }

  // --- normalize and store to AO[B,S,D] ---
#pragma unroll
  for (int r = 0; r < 8; ++r) {
    const int qi = q0 + r + 8 * lh;
    const float inv = 1.0f / lrow[r];
    float* orow = AO + ((size_t)(b * S_LEN + qi)) * DMODEL + h * DH + ln;
    orow[0]  = o0[r] * inv;
    orow[16] = o1[r] * inv;
    orow[32] = o2[r] * inv;
    orow[48] = o3[r] * inv;
  }
}

extern "C" void kernel_launch(void* const* d_in, const int* in_sizes, int n_in,
                              void* d_out, int out_size, void* d_ws, size_t ws_size,
                              hipStream_t stream) {
  const float* X    = (const float*)d_in[0];
  const float* cosT = (const float*)d_in[1];
  const float* sinT = (const float*)d_in[2];
  // d_in[3] = attention_mask (causal, applied analytically in-kernel)
  const float* Wq = (const float*)d_in[4];
  const float* Wk = (const float*)d_in[5];
  const float* Wv = (const float*)d_in[6];
  const float* Wo = (const float*)d_in[7];
  float* out = (float*)d_out;
  float* ws  = (float*)d_ws;

  const size_t QKN = (size_t)BATCH * NH * S_LEN * DH;  // 4M floats each
  float* Qp = ws;
  float* Kp = ws + QKN;
  float* Vp = ws + 2 * QKN;
  float* AO = ws + 3 * QKN;

  dim3 ggrid(DMODEL / 128, (BATCH * S_LEN) / 64);
  gemm_kernel<<<ggrid, 128, 0, stream>>>(X, Wq, Qp, 1);
  gemm_kernel<<<ggrid, 128, 0, stream>>>(X, Wk, Kp, 1);
  gemm_kernel<<<ggrid, 128, 0, stream>>>(X, Wv, Vp, 1);

  const int rope_threads = 2 * BATCH * NH * S_LEN * 32;
  rope_kernel<<<rope_threads / 256, 256, 0, stream>>>(Qp, Kp, cosT, sinT);

  attn_kernel<<<dim3(S_LEN / 128, BATCH * NH), 256, 0, stream>>>(Qp, Kp, Vp, AO);

  gemm_kernel<<<ggrid, 128, 0, stream>>>(AO, Wo, out, 0);
}